// EPG_58025008169014
// MI455X (gfx1250) — compile-verified
//
#include <hip/hip_runtime.h>
#include <math.h>
#include <stdint.h>

// EPG-TSE per-pixel simulation for MI455X (gfx1250, wave32).
// One thread = one pixel; all state floats live in VGPRs. The echo train for
// the benchmark config (nrefocus=16 -> np2=17) is specialized at compile time
// so all truncation bounds fold away -> straight-line f32 FMA code.
// Inputs staged via the gfx1250 async global->LDS path (ASYNCcnt) + prefetch.
// 64-thread blocks: 256 workgroups spread the 512 waves across ~4x more WGPs
// than 256-thread blocks would, minimizing the serialized-chain wall time for
// this latency-bound workload.
//
// WMMA deliberately absent: the only shared operator is a 3x3 rotation
// (kron(I,A)); on v_wmma_f32_16x16x4_f32 that is <=9/64 MAC utilization plus
// per-echo cross-lane marshaling exceeding the 9 FMAs/block it would replace,
// and WMMA cannot predicate the per-echo truncation (EXEC must be all-1s).

#define EPG_BLOCK  64
#define EPG_MAXB   20    // kmax+1 for np2 <= 17
#define EPG_MAXNP2 17
#define EPG_PI     3.14159265358979323846f

#ifndef EPG_TRY_ASYNC
#define EPG_TRY_ASYNC 1
#endif

#if EPG_TRY_ASYNC && defined(__HIP_DEVICE_COMPILE__) && defined(__gfx1250__) && \
    __has_builtin(__builtin_amdgcn_global_load_async_to_lds_b32) &&             \
    __has_builtin(__builtin_amdgcn_s_wait_asynccnt)
#define EPG_ASYNC_LDS 1
#else
#define EPG_ASYNC_LDS 0
#endif

// Refocus: per active block, shared 3x3 rotation (9 FMAs/block).
template<int NB>
__device__ __forceinline__ void epg_refocus(float (&Fp)[NB], float (&Fm)[NB], float (&Z)[NB],
                                            int KB, float c2r, float s2r, float sr,
                                            float hsr, float cr)
{
#pragma unroll
  for (int g = 0; g < NB; ++g) {
    if (g < KB) {
      const float a = Fp[g], b = Fm[g], zz = Z[g];
      Fp[g] = c2r * a + s2r * b + sr * zz;
      Fm[g] = s2r * a + c2r * b - sr * zz;
      Z[g]  = hsr * b - hsr * a + cr * zz;     // -0.5*sr*a + 0.5*sr*b + cr*z
    }
  }
}

// Single half-echo relaxation+shift: x' = E * (S x) + b, truncated to KB blocks.
template<int NB>
__device__ __forceinline__ void epg_half_relax(float (&Fp)[NB], float (&Fm)[NB], float (&Z)[NB],
                                               int KB, float E1, float E2, float b2)
{
  const float t = Fm[1];
#pragma unroll
  for (int g = NB - 1; g >= 1; --g) {          // descending: reads pre-overwrite
    if (g < KB) Fp[g] = E2 * Fp[g - 1];
  }
  Fp[0] = E2 * t;
  Fm[0] = E2 * t;
#pragma unroll
  for (int g = 1; g < NB; ++g) {               // ascending: reads pre-overwrite
    if (g < KB - 1)       Fm[g] = E2 * Fm[g + 1];
    else if (g == KB - 1) Fm[g] = 0.0f;
  }
#pragma unroll
  for (int g = 0; g < NB; ++g) {
    if (g < KB) Z[g] *= E1;
  }
  Z[0] += b2;
}

// Fused double half-relax (both applications share KB):
//   F+0'' = F-0'' = E2^2*F-2,  F+1'' = E2^2*F-1,  F+g'' = E2^2*F+(g-2)  (g>=2)
//   F-g'' = E2^2*F-(g+2) (g<=KB-3),  F-(KB-2)'' = F-(KB-1)'' = 0
//   Zg''  = E1^2*Zg,  Z0'' += (1+E1)*(1-E1)
template<int NB>
__device__ __forceinline__ void epg_relax2(float (&Fp)[NB], float (&Fm)[NB], float (&Z)[NB],
                                           int KB, float EE1, float EE2, float bb2)
{
  const float t1 = Fm[1];
  const float t2 = Fm[2];
#pragma unroll
  for (int g = NB - 1; g >= 2; --g) {          // descending: reads pre-overwrite
    if (g < KB) Fp[g] = EE2 * Fp[g - 2];
  }
  Fp[1] = EE2 * t1;
  Fp[0] = EE2 * t2;
  Fm[0] = EE2 * t2;
#pragma unroll
  for (int g = 1; g < NB; ++g) {               // ascending: reads pre-overwrite
    if (g < KB - 2)                      Fm[g] = EE2 * Fm[g + 2];
    else if (g == KB - 2 || g == KB - 1) Fm[g] = 0.0f;
  }
#pragma unroll
  for (int g = 0; g < NB; ++g) {
    if (g < KB) Z[g] *= EE1;
  }
  Z[0] += bb2;
}

// Compile-time-specialized echo train: all KB values fold to constants.
template<int NP2>
__device__ __forceinline__ float epg_sim(float PD, int Ntr, int inversion, float dInv,
                                         float E1, float E2, float b2, float Xi,
                                         float se, float ce,
                                         float c2r, float s2r, float sr, float cr,
                                         int rec_jj)
{
  constexpr int CH   = (NP2 + 1) / 2;
  constexpr int KRAW = 2 * CH + 1;
  constexpr int KM0  = 2 * (NP2 - 1);
  constexpr int KMAX = (KRAW < KM0) ? KRAW : KM0;
  constexpr int NB   = KMAX + 1;

  const float EE1 = E1 * E1, EE2 = E2 * E2, bb2 = (1.0f + E1) * b2;
  const float hsr = 0.5f * sr;

  float Fp[NB], Fm[NB], Z[NB];
  float z0 = PD, sOut = 0.0f;

  for (int trep = 0; trep < Ntr; ++trep) {
#pragma unroll
    for (int g = 0; g < NB; ++g) { Fp[g] = 0.0f; Fm[g] = 0.0f; Z[g] = 0.0f; }

    float zi = z0;
    if (inversion) zi = dInv * (-zi) + 1.0f - dInv;

    // excitation on block 0 (state was [0,0,zi])
    Fp[0] = se * zi; Fm[0] = se * zi; Z[0] = ce * zi;
    epg_half_relax(Fp, Fm, Z, 2, E1, E2, b2);        // FF[:6] = ES[:6,:6] FF[:6] + b[:6]

    float zn_last = 0.0f;
#pragma unroll
    for (int jj = 1; jj < NP2; ++jj) {
      const int kp = (jj < CH) ? (2 * jj + 3) : (2 * (NP2 - jj) + 1);
      const int KB = (kp > KMAX) ? KMAX : kp;        // folds to a constant per jj

      epg_refocus(Fp, Fm, Z, KB, c2r, s2r, sr, hsr, cr);  // ff = T[:K,:K] FF[:K]

      // fcol values are cheap functions of the post-refocus state:
      //   fcol.F+0 = E2*F-1,  fcol.Z0 = E1*Z0 + (1-E1)
      if (trep == Ntr - 1 && jj == rec_jj) sOut = E2 * Fm[1];
      if (jj == NP2 - 1) {
        zn_last = E1 * Z[0] + b2;                    // Zn[:,0,-1]; reference breaks here
      } else {
        epg_relax2(Fp, Fm, Z, KB, EE1, EE2, bb2);    // FF = ES*(ES*ff + b) + b
      }
    }
    z0 = Xi * zn_last + 1.0f - Xi;
  }
  return fabsf(sOut) * PD;
}

// Generic runtime-np2 fallback (covers nrefocus==1 and any non-benchmark config).
__device__ __noinline__ float epg_sim_dyn(int np2, float PD, int Ntr, int inversion, float dInv,
                                          float E1, float E2, float b2, float Xi,
                                          float se, float ce,
                                          float c2r, float s2r, float sr, float cr,
                                          int rec_jj)
{
  const int CH = (np2 + 1) >> 1;
  int kmax = 2 * (np2 - 1);
  { const int kraw = 2 * CH + 1; if (kraw < kmax) kmax = kraw; }
  const float hsr = 0.5f * sr;

  float Fp[EPG_MAXB], Fm[EPG_MAXB], Z[EPG_MAXB];
  float z0 = PD, sOut = 0.0f;

  for (int trep = 0; trep < Ntr; ++trep) {
#pragma unroll
    for (int g = 0; g < EPG_MAXB; ++g) { Fp[g] = 0.0f; Fm[g] = 0.0f; Z[g] = 0.0f; }

    float zi = z0;
    if (inversion) zi = dInv * (-zi) + 1.0f - dInv;
    Fp[0] = se * zi; Fm[0] = se * zi; Z[0] = ce * zi;
    epg_half_relax(Fp, Fm, Z, 2, E1, E2, b2);

    float zn_last = 0.0f;
    for (int jj = 1; jj < np2; ++jj) {
      int kp = (jj < CH) ? (2 * jj + 3) : (2 * (np2 - jj) + 1);
      if (kp > kmax) kp = kmax;
      const int KB = kp;

      epg_refocus(Fp, Fm, Z, KB, c2r, s2r, sr, hsr, cr);
      epg_half_relax(Fp, Fm, Z, KB, E1, E2, b2);           // -> fcol
      if (trep == Ntr - 1 && jj == rec_jj) sOut = Fp[0];
      if (jj == np2 - 1) { zn_last = Z[0]; break; }
      epg_half_relax(Fp, Fm, Z, KB, E1, E2, b2);           // -> next FF
    }
    z0 = Xi * zn_last + 1.0f - Xi;
  }
  return fabsf(sOut) * PD;
}

__global__ void __launch_bounds__(EPG_BLOCK)
epg_tse_kernel(const float* __restrict__ fake_param,
               const int* __restrict__ p_nref, const int* __restrict__ p_esp,
               const int* __restrict__ p_te,   const int* __restrict__ p_tr,
               const int* __restrict__ p_inv,  const int* __restrict__ p_ti,
               float* __restrict__ out, int im)
{
  const int tid = (int)threadIdx.x;
  const int gid = (int)blockIdx.x * EPG_BLOCK + tid;
  const int lg  = (gid < im) ? gid : (im - 1);   // clamp so barriers stay uniform

  // gfx1250 prefetch path (global_prefetch_b8)
  __builtin_prefetch(fake_param + lg, 0, 1);
  __builtin_prefetch(fake_param + im + lg, 0, 1);
  __builtin_prefetch(fake_param + 2 * im + lg, 0, 1);

  __shared__ float sP[3 * EPG_BLOCK];
#if EPG_ASYNC_LDS
  {
    // gfx1250 async global->LDS path; tracked by ASYNCcnt.
    __attribute__((address_space(1))) int* g1 =
        (__attribute__((address_space(1))) int*)const_cast<float*>(fake_param);
    __attribute__((address_space(3))) int* l3 =
        (__attribute__((address_space(3))) int*)(&sP[0]);
    __builtin_amdgcn_global_load_async_to_lds_b32(&g1[lg],          &l3[tid],                 0, 0);
    __builtin_amdgcn_global_load_async_to_lds_b32(&g1[im + lg],     &l3[EPG_BLOCK + tid],     0, 0);
    __builtin_amdgcn_global_load_async_to_lds_b32(&g1[2 * im + lg], &l3[2 * EPG_BLOCK + tid], 0, 0);
    __builtin_amdgcn_s_wait_asynccnt(0);
  }
  __syncthreads();
#else
  sP[tid]                 = fake_param[lg];
  sP[EPG_BLOCK + tid]     = fake_param[im + lg];
  sP[2 * EPG_BLOCK + tid] = fake_param[2 * im + lg];
  __syncthreads();
#endif

  const float PD = sP[tid];
  const float T1 = sP[EPG_BLOCK + tid]     * 5000.0f + 0.01f;
  const float T2 = sP[2 * EPG_BLOCK + tid] * 1500.0f + 0.01f;

  const int   nrefocus  = p_nref[0];
  const int   iESP      = p_esp[0];
  const int   iTE       = p_te[0];
  const int   inversion = p_inv[0];
  const float fESP      = (float)iESP;
  const float fTR       = (float)p_tr[0];
  const float fTI       = (float)p_ti[0];

  float a_ex, a_ref; int np2, Ntr;
  if (nrefocus == 1) {
    a_ex = 70.0f * EPG_PI / 180.0f; a_ref = EPG_PI;                   Ntr = 5; np2 = 2;
  } else {
    a_ex = 0.5f * EPG_PI;           a_ref = 120.0f * EPG_PI / 180.0f; Ntr = 2; np2 = nrefocus + 1;
  }
  if (np2 > EPG_MAXNP2) np2 = EPG_MAXNP2;   // register-array capacity guard

  const float E1 = expf(-0.5f * fESP / T1);
  const float E2 = expf(-0.5f * fESP / T2);
  const float b2 = 1.0f - E1;
  const float Xi = expf(-(fTR - fESP * ((float)nrefocus + 0.5f)) / T1);
  const float dInv = expf(-fTI / T1);

  const float se = sinf(a_ex), ce = cosf(a_ex);
  const float sr = sinf(a_ref), cr = cosf(a_ref);
  float c2r = cosf(0.5f * a_ref); c2r *= c2r;
  float s2r = sinf(0.5f * a_ref); s2r *= s2r;

  const int rec_jj = (iTE + iESP - 1) / iESP;   // = ceil(TE/ESP) = TE-echo index + 1

  float result;
  if (np2 == 17) {   // nrefocus = 16 (the benchmark config): straight-line code
    result = epg_sim<17>(PD, Ntr, inversion, dInv, E1, E2, b2, Xi,
                         se, ce, c2r, s2r, sr, cr, rec_jj);
  } else {           // faithful generic fallback (incl. nrefocus==1)
    result = epg_sim_dyn(np2, PD, Ntr, inversion, dInv, E1, E2, b2, Xi,
                         se, ce, c2r, s2r, sr, cr, rec_jj);
  }

  if (gid < im) out[gid] = result;
}

extern "C" void kernel_launch(void* const* d_in, const int* in_sizes, int n_in,
                              void* d_out, int out_size, void* d_ws, size_t ws_size,
                              hipStream_t stream)
{
  (void)n_in; (void)out_size; (void)d_ws; (void)ws_size;
  const float* fp   = (const float*)d_in[0];
  const int*   nref = (const int*)d_in[1];
  const int*   esp  = (const int*)d_in[2];
  const int*   te   = (const int*)d_in[3];
  const int*   tr   = (const int*)d_in[4];
  const int*   inv  = (const int*)d_in[5];
  const int*   ti   = (const int*)d_in[6];

  const int im     = in_sizes[0] / 3;           // fake_param is (1,3,im,1) flat
  const int blocks = (im + EPG_BLOCK - 1) / EPG_BLOCK;

  epg_tse_kernel<<<blocks, EPG_BLOCK, 0, stream>>>(fp, nref, esp, te, tr, inv, ti,
                                                   (float*)d_out, im);
}